// LossAF_84189948936764
// MI455X (gfx1250) — compile-verified
//
#include <hip/hip_runtime.h>
#include <math.h>

// ---------------- constants ----------------
#define BATCH   16
#define NGT     60
#define KTOP    10
#define NCAND   (NGT * KTOP)     // 600
#define A0      6400
#define A1      1600
#define A2      400
#define ATOT    8400
#define CCH     80
#define ECH     84
#define NPAD    64               // padded GT columns for WMMA tiles
#define EPSF    1e-7f
#define PIF     3.14159265358979323846f
#define LBW     7.5f
#define LCW     0.5f
#define ACOST   0.5f
#define BCOST   6.0f
#define ALPHAF  0.25f

typedef float v2f __attribute__((ext_vector_type(2)));
typedef float v8f __attribute__((ext_vector_type(8)));

__device__ __forceinline__ float sigmoidf_(float x) { return 1.0f / (1.0f + __expf(-x)); }
__device__ __forceinline__ float softplusf_(float x) { return fmaxf(x, 0.0f) + log1pf(__expf(-fabsf(x))); }

// focal term with tgt = 0 : a_t = 0.75, ce = max(x,0)+log1p(e^-|x|), (1-p_t)^2 = pr^2
__device__ __forceinline__ float focal0(float x) {
  float pr = sigmoidf_(x);
  float ce = fmaxf(x, 0.0f) + log1pf(__expf(-fabsf(x)));
  return 0.75f * ce * pr * pr;
}
// focal term with tgt = 1 : a_t = 0.25, ce = max(x,0)-x+log1p(e^-|x|), (1-p_t)^2=(1-pr)^2
__device__ __forceinline__ float focal1(float x) {
  float pr = sigmoidf_(x);
  float ce = fmaxf(x, 0.0f) - x + log1pf(__expf(-fabsf(x)));
  float om = 1.0f - pr;
  return 0.25f * ce * om * om;
}

__device__ __forceinline__ float fetch_logit(const float* __restrict__ p0,
                                             const float* __restrict__ p1,
                                             const float* __restrict__ p2,
                                             int b, int a, int ch) {
  if (a < A0)        return p0[(size_t)(b * A0 + a) * ECH + 4 + ch];
  if (a < A0 + A1)   return p1[(size_t)(b * A1 + (a - A0)) * ECH + 4 + ch];
  return p2[(size_t)(b * A2 + (a - A0 - A1)) * ECH + 4 + ch];
}

// lexicographic (val, idx) wave32 argmin
__device__ __forceinline__ void wave_argmin(float& v, int& i) {
#pragma unroll
  for (int off = 16; off > 0; off >>= 1) {
    float ov = __shfl_xor(v, off, 32);
    int   oi = __shfl_xor(i, off, 32);
    if (ov < v || (ov == v && oi < i)) { v = ov; i = oi; }
  }
}

// ---------------- kernels ----------------
__global__ void init_k(float* out, float* base) {
  if (threadIdx.x == 0) out[0] = 0.0f;
  if (threadIdx.x < BATCH) base[threadIdx.x] = 0.0f;
}

// decode anchors -> boxes[b][a] = (px,py,pw,ph)
__global__ void decode_k(const float* __restrict__ p0, const float* __restrict__ p1,
                         const float* __restrict__ p2, float* __restrict__ boxes) {
  int idx = blockIdx.x * blockDim.x + threadIdx.x;
  if (idx >= BATCH * ATOT) return;
  int b = idx / ATOT, a = idx - b * ATOT;
  const float* src; int S, li; float stride;
  if (a < A0)          { li = a;            src = p0 + (size_t)(b * A0 + li) * ECH; S = 80; stride = 8.0f;  }
  else if (a < A0+A1)  { li = a - A0;       src = p1 + (size_t)(b * A1 + li) * ECH; S = 40; stride = 16.0f; }
  else                 { li = a - A0 - A1;  src = p2 + (size_t)(b * A2 + li) * ECH; S = 20; stride = 32.0f; }
  int gy = li / S, gx = li - gy * S;
  float4 t = *(const float4*)src;
  float px = (sigmoidf_(t.x) * 2.0f - 0.5f + (float)gx) * stride;
  float py = (sigmoidf_(t.y) * 2.0f - 0.5f + (float)gy) * stride;
  float pw = softplusf_(t.z) * stride;
  float ph = softplusf_(t.w) * stride;
  *(float4*)(boxes + (size_t)idx * 4) = make_float4(px, py, pw, ph);
}

// streaming focal-loss baseline (tgt==0 everywhere), per-image sum into base[b]
__global__ void focal_base_k(const float* __restrict__ p, int Alev, float* __restrict__ base) {
  int b = blockIdx.y;
  int t = blockIdx.x * blockDim.x + threadIdx.x;  // float4 group within image-level
  int ngroups = Alev * (CCH / 4);                 // 20 float4 groups per anchor
  float s = 0.0f;
  if (t < ngroups) {
    int aIdx = t / 20, g = t - aIdx * 20;
    const float* ptr = p + (size_t)(b * Alev + aIdx) * ECH + 4 + g * 4;
    __builtin_prefetch(ptr + 84 * 16);            // global_prefetch_b8 (speculative)
    float4 v = *(const float4*)ptr;
    s = focal0(v.x) + focal0(v.y) + focal0(v.z) + focal0(v.w);
  }
  __shared__ float red[256];
  red[threadIdx.x] = s;
  __syncthreads();
  for (int st = 128; st > 0; st >>= 1) {
    if (threadIdx.x < st) red[threadIdx.x] += red[threadIdx.x + st];
    __syncthreads();
  }
  if (threadIdx.x == 0) atomicAdd(&base[b], red[0]);
}

// squared-distance matrix via rank-4 GEMM: dist[b][n][a] = (px-tcx)^2 + (py-tcy)^2
// A row  = [px, py, px^2+py^2, 1]   (16 anchors x K=4)
// B col  = [-2tcx, -2tcy, 1, tcx^2+tcy^2]   (K=4 x 16 GTs)
__global__ void dist_wmma_k(const float* __restrict__ boxes, const float* __restrict__ gt,
                            float* __restrict__ dist) {
  int mTile = blockIdx.x;          // 0..524  (anchors/16)
  int nTile = blockIdx.y;          // 0..3    (padded GT cols/16)
  int b     = blockIdx.z;
  int l  = threadIdx.x;            // wave32
  int lo = l & 15, hi = l >> 4;

  int m = mTile * 16 + lo;
  float2 pc = *(const float2*)(boxes + (size_t)(b * ATOT + m) * 4);   // px, py
  int g = nTile * 16 + lo;
  int gc = g < NGT ? g : NGT - 1;                                     // clamp (pad cols unused)
  const float* gp = gt + (size_t)(b * NGT + gc) * 4;
  float tcx = 0.5f * (gp[0] + gp[2]);
  float tcy = 0.5f * (gp[1] + gp[3]);

  // ISA A-layout: lanes 0-15 hold K=0,1 ; lanes 16-31 hold K=2,3
  v2f A, Bv;
  A.x  = hi ? (pc.x * pc.x + pc.y * pc.y) : pc.x;
  A.y  = hi ? 1.0f                         : pc.y;
  Bv.x = hi ? 1.0f                         : (-2.0f * tcx);
  Bv.y = hi ? (tcx * tcx + tcy * tcy)      : (-2.0f * tcy);

  v8f c = {};
  v8f d = __builtin_amdgcn_wmma_f32_16x16x4_f32(false, A, false, Bv, (short)0, c, false, false);

  // D layout: VGPR r, lane -> row = r + 8*hi, col = lo. 8 consecutive rows per lane.
  int n = nTile * 16 + lo;                                   // < NPAD always
  float* dst = dist + ((size_t)(b * NPAD + n)) * ATOT + mTile * 16 + hi * 8;
  *(float4*)(dst)     = make_float4(d[0], d[1], d[2], d[3]);
  *(float4*)(dst + 4) = make_float4(d[4], d[5], d[6], d[7]);
}

// per (image, GT): top-10 smallest distances -> cand[b][gt][k]
__global__ void topk_k(const float* __restrict__ dist, int* __restrict__ cand) {
  int gtid = blockIdx.x, b = blockIdx.y, l = threadIdx.x;
  const float* row = dist + ((size_t)(b * NPAD + gtid)) * ATOT;
  float bd[KTOP]; int bi[KTOP];
#pragma unroll
  for (int k = 0; k < KTOP; ++k) { bd[k] = 1e30f; bi[k] = 0x7fffffff; }
  for (int a = l; a < ATOT; a += 32) {
    float v = row[a];
    if (v < bd[KTOP - 1] || (v == bd[KTOP - 1] && a < bi[KTOP - 1])) {
      float cv = v; int ci = a;
#pragma unroll
      for (int k = 0; k < KTOP; ++k) {
        bool sw = (cv < bd[k]) || (cv == bd[k] && ci < bi[k]);
        float tv = bd[k]; int ti = bi[k];
        if (sw) { bd[k] = cv; bi[k] = ci; cv = tv; ci = ti; }
      }
    }
  }
  // 10 selection rounds: global min of per-lane heads (lists are sorted)
  for (int r = 0; r < KTOP; ++r) {
    float mv = bd[0]; int mi = bi[0];
    wave_argmin(mv, mi);
    bool winner = (bi[0] == mi) && (bd[0] == mv);   // anchor indices unique across lanes
    if (winner) {
      cand[(size_t)(b * NGT + gtid) * KTOP + r] = mi;
#pragma unroll
      for (int k = 0; k < KTOP - 1; ++k) { bd[k] = bd[k + 1]; bi[k] = bi[k + 1]; }
      bd[KTOP - 1] = 1e30f; bi[KTOP - 1] = 0x7fffffff;
    }
  }
}

// per (image, GT): argmin over the 600-candidate pool of cost = 0.5(1-cls)+6(1-iou)
__global__ void match_k(const float* __restrict__ p0, const float* __restrict__ p1,
                        const float* __restrict__ p2, const float* __restrict__ boxes,
                        const float* __restrict__ gt, const int* __restrict__ labels,
                        const int* __restrict__ cand, int* __restrict__ matched) {
  int j = blockIdx.x, b = blockIdx.y, l = threadIdx.x;
  const float* gp = gt + (size_t)(b * NGT + j) * 4;
  float gx1 = gp[0], gy1 = gp[1], gx2 = gp[2], gy2 = gp[3];
  float area2 = fmaxf(gx2 - gx1, 0.0f) * fmaxf(gy2 - gy1, 0.0f);
  int lab = labels[b * NGT + j];

  float bc = 1e30f; int bidx = 0x7fffffff;
  for (int i = l; i < NCAND; i += 32) {
    int a = cand[(size_t)b * NCAND + i];
    float4 bx = *(const float4*)(boxes + (size_t)(b * ATOT + a) * 4);
    float px1 = bx.x - bx.z * 0.5f, py1 = bx.y - bx.w * 0.5f;
    float px2 = bx.x + bx.z * 0.5f, py2 = bx.y + bx.w * 0.5f;
    float area1 = fmaxf(px2 - px1, 0.0f) * fmaxf(py2 - py1, 0.0f);
    float inter = fmaxf(fminf(px2, gx2) - fmaxf(px1, gx1), 0.0f) *
                  fmaxf(fminf(py2, gy2) - fmaxf(py1, gy1), 0.0f);
    float uni = area1 + area2 - inter + EPSF;
    float iou = fminf(fmaxf(inter / uni, 0.0f), 1.0f);
    float cls = sigmoidf_(fetch_logit(p0, p1, p2, b, a, lab));
    float cost = ACOST * (1.0f - cls) + BCOST * (1.0f - iou);
    if (cost < bc || (cost == bc && i < bidx)) { bc = cost; bidx = i; }
  }
  wave_argmin(bc, bidx);
  if (l == 0) matched[b * NGT + j] = cand[(size_t)b * NCAND + bidx];
}

// per image: CIoU box loss + focal correction (tgt 0 -> 1 at matched, dedup) + combine
__global__ void final_k(const float* __restrict__ p0, const float* __restrict__ p1,
                        const float* __restrict__ p2, const float* __restrict__ boxes,
                        const float* __restrict__ gt, const int* __restrict__ labels,
                        const int* __restrict__ matched, const float* __restrict__ base,
                        float* __restrict__ out) {
  int b = blockIdx.x, j = threadIdx.x;
  __shared__ int sm[NGT], sl[NGT];
  __shared__ float red[64];
  if (j < NGT) { sm[j] = matched[b * NGT + j]; sl[j] = labels[b * NGT + j]; }
  __syncthreads();

  float boxterm = 0.0f, corrterm = 0.0f;
  if (j < NGT) {
    int a = sm[j], lab = sl[j];
    float4 bx = *(const float4*)(boxes + (size_t)(b * ATOT + a) * 4);
    float px1 = bx.x - bx.z * 0.5f, py1 = bx.y - bx.w * 0.5f;
    float px2 = bx.x + bx.z * 0.5f, py2 = bx.y + bx.w * 0.5f;
    const float* gp = gt + (size_t)(b * NGT + j) * 4;
    float tx1 = gp[0], ty1 = gp[1], tx2 = gp[2], ty2 = gp[3];
    float pw = fmaxf(px2 - px1, EPSF), ph = fmaxf(py2 - py1, EPSF);
    float tw = fmaxf(tx2 - tx1, EPSF), th = fmaxf(ty2 - ty1, EPSF);
    float inter = fmaxf(fminf(px2, tx2) - fmaxf(px1, tx1), 0.0f) *
                  fmaxf(fminf(py2, ty2) - fmaxf(py1, ty1), 0.0f);
    float uni = pw * ph + tw * th - inter + EPSF;
    float iou = inter / uni;
    float dx = 0.5f * (px1 + px2) - 0.5f * (tx1 + tx2);
    float dy = 0.5f * (py1 + py2) - 0.5f * (ty1 + ty2);
    float cd = dx * dx + dy * dy;
    float cw = fmaxf(px2, tx2) - fminf(px1, tx1);
    float ch = fmaxf(py2, ty2) - fminf(py1, ty1);
    float c2 = cw * cw + ch * ch + EPSF;
    float da = atanf(tw / th) - atanf(pw / ph);
    float v = (4.0f / (PIF * PIF)) * da * da;
    float alpha = v / (v - iou + 1.0f + EPSF);
    float ciou = iou - cd / c2 - alpha * v;
    boxterm = 1.0f - ciou;

    float x = fetch_logit(p0, p1, p2, b, a, lab);
    bool dup = false;
    for (int jj = 0; jj < j; ++jj)
      if (sm[jj] == a && sl[jj] == lab) dup = true;
    corrterm = dup ? 0.0f : (focal1(x) - focal0(x));
  }

  red[j] = boxterm; __syncthreads();
  for (int st = 32; st > 0; st >>= 1) { if (j < st) red[j] += red[j + st]; __syncthreads(); }
  float sbox = red[0]; __syncthreads();
  red[j] = corrterm; __syncthreads();
  for (int st = 32; st > 0; st >>= 1) { if (j < st) red[j] += red[j + st]; __syncthreads(); }
  if (j == 0) {
    float loss = LBW * sbox / (float)NGT + LCW * (base[b] + red[0]) / (float)NGT;
    atomicAdd(out, loss / (float)BATCH);
  }
}

// ---------------- launch ----------------
extern "C" void kernel_launch(void* const* d_in, const int* in_sizes, int n_in,
                              void* d_out, int out_size, void* d_ws, size_t ws_size,
                              hipStream_t stream) {
  (void)in_sizes; (void)n_in; (void)out_size; (void)ws_size;
  const float* p0 = (const float*)d_in[0];
  const float* p1 = (const float*)d_in[1];
  const float* p2 = (const float*)d_in[2];
  const float* gt = (const float*)d_in[3];
  const int* labels = (const int*)d_in[4];
  float* out = (float*)d_out;

  float* boxes  = (float*)d_ws;                       // 16*8400*4 floats
  float* dist   = boxes + (size_t)BATCH * ATOT * 4;   // 16*64*8400 floats (L2-resident)
  float* base   = dist + (size_t)BATCH * NPAD * ATOT; // 16 floats
  int*   cand   = (int*)(base + BATCH);               // 16*600 ints
  int*   match  = cand + BATCH * NCAND;               // 16*60 ints

  init_k<<<1, 32, 0, stream>>>(out, base);
  decode_k<<<(BATCH * ATOT + 255) / 256, 256, 0, stream>>>(p0, p1, p2, boxes);
  focal_base_k<<<dim3((A0 * 20 + 255) / 256, BATCH), 256, 0, stream>>>(p0, A0, base);
  focal_base_k<<<dim3((A1 * 20 + 255) / 256, BATCH), 256, 0, stream>>>(p1, A1, base);
  focal_base_k<<<dim3((A2 * 20 + 255) / 256, BATCH), 256, 0, stream>>>(p2, A2, base);
  dist_wmma_k<<<dim3(ATOT / 16, NPAD / 16, BATCH), 32, 0, stream>>>(boxes, gt, dist);
  topk_k<<<dim3(NGT, BATCH), 32, 0, stream>>>(dist, cand);
  match_k<<<dim3(NGT, BATCH), 32, 0, stream>>>(p0, p1, p2, boxes, gt, labels, cand, match);
  final_k<<<BATCH, 64, 0, stream>>>(p0, p1, p2, boxes, gt, labels, match, base, out);
}